// SplineInterpolation_ND_BCXYZ_18176301597165
// MI455X (gfx1250) — compile-verified
//
#include <hip/hip_runtime.h>
#include <stdint.h>

// ---------------------------------------------------------------------------
// 3D spline interpolation, B=2, C=2, K=4 taps/axis, 128^3 grid.
// Memory-bound: weight+index streams (~402MB) dominate; c (33.5MB) is
// L2-resident on MI455X (192MB L2). Strategy:
//   - TDM (tensor_load_to_lds) double-buffered staging of weight/index tiles
//     into LDS, overlapped with compute (TENSORcnt async path).
//   - Non-temporal output stores to protect L2 residency of c.
//   - Default-policy 4B gathers into c (served by L2).
// ---------------------------------------------------------------------------

#define TILE   256
#define NGRID  128
#define NVOX   (NGRID * NGRID * NGRID)   // 2097152 voxels per batch
#define NBATCH 2
#define NCHAN  2
#define KTAPS  4

typedef uint32_t u32;
typedef uint64_t u64;
typedef u32 v4u __attribute__((ext_vector_type(4)));
typedef int v8i __attribute__((ext_vector_type(8)));
typedef int v4i __attribute__((ext_vector_type(4)));

// Issue one TDM tensor load: 3D tile [tile0 x tile1 x tile2] of 4-byte
// elements, row stride = stride0 elems, plane stride = stride1 elems,
// packed contiguously into LDS at lds_addr.
__device__ __forceinline__ void tdm_load_3d(u64 gaddr, u32 lds_addr,
                                            u32 tile0, u32 tile1, u32 tile2,
                                            u64 stride0, u64 stride1) {
  v4u g0;
  g0.x = 0x1u;                                   // count=1, user descriptor
  g0.y = lds_addr;                               // lds_addr[31:0]
  g0.z = (u32)(gaddr & 0xffffffffu);             // global_addr[31:0]
  g0.w = (u32)((gaddr >> 32) & 0x01ffffffu)      // global_addr[56:32]
         | (2u << 30);                           // type = 2 ("image")

  v8i g1;
  g1[0] = (int)(2u << 16);                       // wg_mask=0, data_size=2 (4B)
  g1[1] = (int)((tile0 & 0xffffu) << 16);        // abar=0 | tensor_dim0[15:0]
  g1[2] = (int)(((tile0 >> 16) & 0xffffu) | ((tile1 & 0xffffu) << 16));
  g1[3] = (int)(((tile1 >> 16) & 0xffffu) | ((tile0 & 0xffffu) << 16)); // tile_dim0
  g1[4] = (int)((tile1 & 0xffffu) | ((tile2 & 0xffffu) << 16));         // tile_dim1,2
  g1[5] = (int)(stride0 & 0xffffffffu);                                  // dim0_stride[31:0]
  g1[6] = (int)(((stride0 >> 32) & 0xffffu) | ((stride1 & 0xffffu) << 16));
  g1[7] = (int)((stride1 >> 16) & 0xffffffffu);                          // dim1_stride[47:16]

  v4i g2;
  g2[0] = (int)tile2;   // tensor_dim2 (OOB bound for the z axis)
  g2[1] = 0;            // tensor_dim3 / lds_addr_increment
  g2[2] = 0;            // tensor_dim2_stride low (tile_dim3 unused)
  g2[3] = 0;            // tensor_dim2_stride hi | tile_dim3=0
  v4i g3 = {0, 0, 0, 0};
  v8i gx = {0, 0, 0, 0, 0, 0, 0, 0};  // extra group (6-arg toolchain), unused

  __builtin_amdgcn_tensor_load_to_lds(g0, g1, g2, g3, gx, 0);
}

__global__ __launch_bounds__(TILE) void spline_interp3d_kernel(
    const float* __restrict__ c,       // [B, C, 128,128,128]
    const float* __restrict__ weight,  // [K, B, 3, 128,128,128]
    const int*   __restrict__ index,   // [K, B, 3, 128,128,128]
    float* __restrict__ out,           // [B, C, 128,128,128]
    int totalTiles) {
  // Double-buffered LDS staging: [buf][k*3+dim][TILE]
  __shared__ float s_w[2][KTAPS * 3 * TILE];   // 24 KB
  __shared__ int   s_i[2][KTAPS * 3 * TILE];   // 24 KB

  const int tilesPerBatch = NVOX / TILE;       // 8192
  const u64 dimStride = (u64)NVOX;             // dim axis stride (elems)
  const u64 kStride   = (u64)(NBATCH * 3) * NVOX;  // k axis stride (elems)

  int g = (int)blockIdx.x;
  if (g >= totalTiles) return;
  const bool issuer = (threadIdx.x < 32);      // wave 0 (wave32)

  // ---- prologue: stage first tile --------------------------------------
  if (issuer) {
    int b  = g / tilesPerBatch;
    int s0 = (g - b * tilesPerBatch) * TILE;
    u64 woff = ((u64)b * 3 * NVOX + (u64)s0) * 4u;
    tdm_load_3d((u64)(uintptr_t)weight + woff, (u32)(uintptr_t)&s_w[0][0],
                TILE, 3, KTAPS, dimStride, kStride);
    tdm_load_3d((u64)(uintptr_t)index + woff, (u32)(uintptr_t)&s_i[0][0],
                TILE, 3, KTAPS, dimStride, kStride);
  }

  int cur = 0;
  while (g < totalTiles) {
    const int nxt = g + (int)gridDim.x;
    const bool has_next = (nxt < totalTiles);

    // ---- issue next tile's DMA, then wait for current tile ------------
    if (issuer) {
      if (has_next) {
        int bn  = nxt / tilesPerBatch;
        int s0n = (nxt - bn * tilesPerBatch) * TILE;
        u64 woff = ((u64)bn * 3 * NVOX + (u64)s0n) * 4u;
        tdm_load_3d((u64)(uintptr_t)weight + woff,
                    (u32)(uintptr_t)&s_w[cur ^ 1][0],
                    TILE, 3, KTAPS, dimStride, kStride);
        tdm_load_3d((u64)(uintptr_t)index + woff,
                    (u32)(uintptr_t)&s_i[cur ^ 1][0],
                    TILE, 3, KTAPS, dimStride, kStride);
        __builtin_amdgcn_s_wait_tensorcnt(2);  // current tile's 2 ops done
      } else {
        __builtin_amdgcn_s_wait_tensorcnt(0);
      }
    }
    __syncthreads();   // staged data visible to all waves

    // ---- compute current tile ------------------------------------------
    {
      const int b  = g / tilesPerBatch;
      const int s0 = (g - b * tilesPerBatch) * TILE;
      const int t  = (int)threadIdx.x;

      float wx[KTAPS], wy[KTAPS], wz[KTAPS];
      int   ox[KTAPS], oy[KTAPS], oz[KTAPS];
#pragma unroll
      for (int k = 0; k < KTAPS; ++k) {
        wx[k] = s_w[cur][(k * 3 + 0) * TILE + t];
        wy[k] = s_w[cur][(k * 3 + 1) * TILE + t];
        wz[k] = s_w[cur][(k * 3 + 2) * TILE + t];
        ox[k] = s_i[cur][(k * 3 + 0) * TILE + t] << 14;  // *128*128
        oy[k] = s_i[cur][(k * 3 + 1) * TILE + t] << 7;   // *128
        oz[k] = s_i[cur][(k * 3 + 2) * TILE + t];
      }

      const float* __restrict__ c0 = c + (size_t)b * NCHAN * NVOX;  // L2-resident
      const float* __restrict__ c1 = c0 + NVOX;

      float acc0 = 0.f, acc1 = 0.f;
#pragma unroll
      for (int a = 0; a < KTAPS; ++a) {
#pragma unroll
        for (int bt = 0; bt < KTAPS; ++bt) {
          const float wab = wx[a] * wy[bt];
          const int   oab = ox[a] + oy[bt];
#pragma unroll
          for (int d = 0; d < KTAPS; ++d) {
            const float w = wab * wz[d];
            const int   o = oab + oz[d];
            acc0 = fmaf(w, c0[o], acc0);
            acc1 = fmaf(w, c1[o], acc1);
          }
        }
      }

      // NT stores: write-once output must not evict c from L2
      const size_t o0 = (size_t)b * NCHAN * NVOX + (size_t)s0 + (size_t)t;
      __builtin_nontemporal_store(acc0, out + o0);
      __builtin_nontemporal_store(acc1, out + o0 + NVOX);
    }

    __syncthreads();   // all waves done reading buf[cur]; safe to overwrite
    cur ^= 1;
    g = nxt;
  }
}

extern "C" void kernel_launch(void* const* d_in, const int* in_sizes, int n_in,
                              void* d_out, int out_size, void* d_ws, size_t ws_size,
                              hipStream_t stream) {
  (void)in_sizes; (void)n_in; (void)d_ws; (void)ws_size; (void)out_size;
  const float* c      = (const float*)d_in[0];
  const float* weight = (const float*)d_in[1];
  const int*   index  = (const int*)d_in[2];
  float*       out    = (float*)d_out;

  const int totalTiles = (NBATCH * NVOX) / TILE;  // 16384
  int blocks = 2048;                              // 8 tiles/block, pipelined
  if (blocks > totalTiles) blocks = totalTiles;

  spline_interp3d_kernel<<<blocks, TILE, 0, stream>>>(c, weight, index, out,
                                                      totalTiles);
}